// SetAbstraction_79061757985222
// MI455X (gfx1250) — compile-verified
//
#include <hip/hip_runtime.h>
#include <hip/hip_bf16.h>

typedef __attribute__((ext_vector_type(2))) float v2f;
typedef __attribute__((ext_vector_type(8))) float v8f;

#define NB      8
#define NPT     16384
#define NPOINT  1024
#define NSAMP   64
#define CIN     64
#define C1      64
#define C2      128
#define RAD2    0.04f
#define XPITCH  66   // LDS row pitch (floats) to dodge bank conflicts

// ---------------------------------------------------------------------------
// Kernel 1: farthest point sampling. One block per batch; 1024 threads hold
// 16 points each in registers. Sequential 1024-step loop (inherent to FPS):
// per step: distance update (16 FMA-chains/thread), wave shfl argmax,
// cross-wave LDS argmax, winner broadcasts its coords via LDS.
// ---------------------------------------------------------------------------
__global__ __launch_bounds__(1024) void fps_kernel(const float* __restrict__ xyz,
                                                   int* __restrict__ idx_out) {
  const int b    = blockIdx.x;
  const int tid  = threadIdx.x;
  const int wv   = tid >> 5;
  const int lane = tid & 31;
  const float* xb = xyz + (size_t)b * NPT * 3;

  float px[16], py[16], pz[16], dd[16];
#pragma unroll
  for (int j = 0; j < 16; ++j) {
    int p = tid + j * 1024;
    px[j] = xb[p * 3 + 0];
    py[j] = xb[p * 3 + 1];
    pz[j] = xb[p * 3 + 2];
    dd[j] = 1e10f;
  }

  __shared__ float s_ptx, s_pty, s_ptz;
  __shared__ float s_rd[32];
  __shared__ int   s_ri[32];
  __shared__ int   s_last;
  if (tid == 0) s_last = 0;
  __syncthreads();

  for (int it = 0; it < NPOINT; ++it) {
    int last = s_last;
    if (tid == 0) idx_out[b * NPOINT + it] = last;
    // owner of `last` publishes its coordinates
    if ((last & 1023) == tid) {
      int slot = last >> 10;
#pragma unroll
      for (int j = 0; j < 16; ++j)
        if (j == slot) { s_ptx = px[j]; s_pty = py[j]; s_ptz = pz[j]; }
    }
    __syncthreads();

    float cx = s_ptx, cy = s_pty, cz = s_ptz;
    float bd = -1.0f; int bi = 0;
#pragma unroll
    for (int j = 0; j < 16; ++j) {
      float dx = px[j] - cx, dy = py[j] - cy, dz = pz[j] - cz;
      float d  = dx * dx + dy * dy + dz * dz;
      float nd = fminf(dd[j], d);
      dd[j] = nd;
      int p = tid + j * 1024;
      if (nd > bd) { bd = nd; bi = p; }   // ascending p: first-max kept
    }
    // wave32 argmax (tie -> smaller index, matching jnp.argmax)
#pragma unroll
    for (int off = 16; off > 0; off >>= 1) {
      float od = __shfl_xor(bd, off, 32);
      int   oi = __shfl_xor(bi, off, 32);
      if (od > bd || (od == bd && oi < bi)) { bd = od; bi = oi; }
    }
    if (lane == 0) { s_rd[wv] = bd; s_ri[wv] = bi; }
    __syncthreads();
    if (wv == 0) {
      float vd = s_rd[lane];
      int   vi = s_ri[lane];
#pragma unroll
      for (int off = 16; off > 0; off >>= 1) {
        float od = __shfl_xor(vd, off, 32);
        int   oi = __shfl_xor(vi, off, 32);
        if (od > vd || (od == vd && oi < vi)) { vd = od; vi = oi; }
      }
      if (lane == 0) s_last = vi;
    }
    __syncthreads();
  }
}

// ---------------------------------------------------------------------------
// Kernel 2: ordered ball query with early exit. One block per centroid.
// Scans points in ascending index order in chunks of 256; wave32 ballot +
// LDS prefix gives in-order ranks; stops once 64 found (~8 chunks typical).
// Mimics reference numerics: d2 = |c|^2 + |p|^2 - 2 c.p ; pads with first
// hit; empty ball pads with NPT-1 (JAX clamp of OOB gather index n).
// ---------------------------------------------------------------------------
__global__ __launch_bounds__(256) void ballq_kernel(const float* __restrict__ xyz,
                                                    const int* __restrict__ fidx,
                                                    int* __restrict__ gidx) {
  const int bs   = blockIdx.x;
  const int b    = bs >> 10;
  const int tid  = threadIdx.x;
  const int wv   = tid >> 5;
  const int lane = tid & 31;
  const float* xb = xyz + (size_t)b * NPT * 3;

  int ci = fidx[bs];
  float cx = xb[ci * 3 + 0], cy = xb[ci * 3 + 1], cz = xb[ci * 3 + 2];
  float cn = cx * cx + cy * cy + cz * cz;

  __shared__ int s_wc[8];
  __shared__ int s_cnt;
  __shared__ int s_sel[NSAMP];
  if (tid == 0) s_cnt = 0;

  int total = 0;
  for (int base = 0; base < NPT; base += 256) {
    int p = base + tid;
    float x = xb[p * 3 + 0], y = xb[p * 3 + 1], z = xb[p * 3 + 2];
    float d2 = cn + (x * x + y * y + z * z) - 2.0f * (cx * x + cy * y + cz * z);
    bool in = d2 <= RAD2;
    unsigned mask = __builtin_amdgcn_ballot_w32(in);
    if (lane == 0) s_wc[wv] = __popc(mask);
    __syncthreads();
    int pre = 0, tot = 0;
#pragma unroll
    for (int w = 0; w < 8; ++w) { int c = s_wc[w]; tot += c; if (w < wv) pre += c; }
    int cnt = s_cnt;
    if (in) {
      int rank = cnt + pre + __popc(mask & ((1u << lane) - 1u));
      if (rank < NSAMP) s_sel[rank] = p;
    }
    total = cnt + tot;
    __syncthreads();
    if (tid == 0) s_cnt = total;
    if (total >= NSAMP) break;   // uniform across block
  }
  __syncthreads();
  if (tid < NSAMP) {
    int v;
    if (total == 0)        v = NPT - 1;          // empty ball -> clamped index
    else if (tid < total)  v = s_sel[tid];
    else                   v = s_sel[0];         // pad with first hit
    gidx[(size_t)bs * NSAMP + tid] = v;
  }
}

// ---------------------------------------------------------------------------
// Kernel 3: fused gather + MLP(64->64) + BN/ReLU + MLP(64->128) + BN/ReLU.
// One block per (b,s) group; 256 threads = 8 waves. X(64x64) gathered to LDS,
// layer-1 result H kept in LDS; all matmuls via exact-precision
// V_WMMA_F32_16X16X4_F32 (memory-bound problem -> f32 WMMA is the right
// precision). Each wave: 2 layer-1 tiles, 4 layer-2 tiles; K-loop = 16 WMMAs
// per 16x16 tile.
// ---------------------------------------------------------------------------
__global__ __launch_bounds__(256) void samlp_kernel(
    const float* __restrict__ points, const int* __restrict__ gidx,
    const float* __restrict__ W0, const float* __restrict__ b0,
    const float* __restrict__ g0, const float* __restrict__ be0,
    const float* __restrict__ m0, const float* __restrict__ v0,
    const float* __restrict__ W1, const float* __restrict__ b1,
    const float* __restrict__ g1, const float* __restrict__ be1,
    const float* __restrict__ m1, const float* __restrict__ v1,
    float* __restrict__ out) {
  const int bs   = blockIdx.x;
  const int b    = bs >> 10;
  const int tid  = threadIdx.x;
  const int wv   = tid >> 5;
  const int lane = tid & 31;

  __shared__ float Xs[64 * XPITCH];
  __shared__ float Hs[64 * XPITCH];

  // ---- gather X[64][64] = points[b, gidx[bs], :] into LDS ----
  {
    int r  = tid >> 2;
    int cb = (tid & 3) * 16;
    int src = gidx[(size_t)bs * NSAMP + r];
    const float* prow = points + ((size_t)b * NPT + src) * CIN;
#pragma unroll
    for (int c = 0; c < 16; ++c) Xs[r * XPITCH + cb + c] = prow[cb + c];
  }
  __syncthreads();

  const int lrow  = lane & 15;
  const int koff  = (lane < 16) ? 0 : 2;   // A/B frag: low lanes K0/K1, high K2/K3
  const int mhalf = (lane < 16) ? 0 : 8;   // C/D frag: low lanes M=j, high M=j+8

  // ---- layer 1: H = relu(bn(X @ W0^T + b0)) ; 16 tiles over 8 waves ----
#pragma unroll
  for (int q = 0; q < 2; ++q) {
    int t   = wv * 2 + q;
    int mt  = (t >> 2) * 16;
    int nt  = (t & 3) * 16;
    int arow = mt + lrow;
    int nch  = nt + lrow;
    const float* wrow = W0 + nch * CIN;   // B[c][o] = W0[o][c]
    v8f acc = {};
#pragma unroll
    for (int k0 = 0; k0 < CIN; k0 += 4) {
      v2f a, bb;
      a.x  = Xs[arow * XPITCH + k0 + koff];
      a.y  = Xs[arow * XPITCH + k0 + koff + 1];
      bb.x = wrow[k0 + koff];
      bb.y = wrow[k0 + koff + 1];
      acc = __builtin_amdgcn_wmma_f32_16x16x4_f32(false, a, false, bb,
                                                  (short)0, acc, false, false);
    }
    float sc = g0[nch] * rsqrtf(v0[nch] + 1e-5f);
    float sh = be0[nch] + (b0[nch] - m0[nch]) * sc;
#pragma unroll
    for (int j = 0; j < 8; ++j) {
      float h = fmaxf(acc[j] * sc + sh, 0.0f);
      Hs[(mt + mhalf + j) * XPITCH + nch] = h;
    }
  }
  __syncthreads();

  // ---- layer 2: out = relu(bn(H @ W1^T + b1)) ; 32 tiles over 8 waves ----
  float* obase = out + (size_t)bs * 64 * C2;
#pragma unroll
  for (int q = 0; q < 4; ++q) {
    int t   = wv * 4 + q;
    int mt  = (t >> 3) * 16;
    int nt  = (t & 7) * 16;
    int arow = mt + lrow;
    int nch  = nt + lrow;
    const float* wrow = W1 + nch * C1;
    v8f acc = {};
#pragma unroll
    for (int k0 = 0; k0 < C1; k0 += 4) {
      v2f a, bb;
      a.x  = Hs[arow * XPITCH + k0 + koff];
      a.y  = Hs[arow * XPITCH + k0 + koff + 1];
      bb.x = wrow[k0 + koff];
      bb.y = wrow[k0 + koff + 1];
      acc = __builtin_amdgcn_wmma_f32_16x16x4_f32(false, a, false, bb,
                                                  (short)0, acc, false, false);
    }
    float sc = g1[nch] * rsqrtf(v1[nch] + 1e-5f);
    float sh = be1[nch] + (b1[nch] - m1[nch]) * sc;
#pragma unroll
    for (int j = 0; j < 8; ++j) {
      float h = fmaxf(acc[j] * sc + sh, 0.0f);
      obase[(size_t)(mt + mhalf + j) * C2 + nch] = h;
    }
  }
}

// ---------------------------------------------------------------------------
extern "C" void kernel_launch(void* const* d_in, const int* in_sizes, int n_in,
                              void* d_out, int out_size, void* d_ws, size_t ws_size,
                              hipStream_t stream) {
  const float* xyz    = (const float*)d_in[0];
  const float* points = (const float*)d_in[1];
  const float* W0  = (const float*)d_in[2];
  const float* b0  = (const float*)d_in[3];
  const float* g0  = (const float*)d_in[4];
  const float* be0 = (const float*)d_in[5];
  const float* m0  = (const float*)d_in[6];
  const float* v0  = (const float*)d_in[7];
  const float* W1  = (const float*)d_in[8];
  const float* b1  = (const float*)d_in[9];
  const float* g1  = (const float*)d_in[10];
  const float* be1 = (const float*)d_in[11];
  const float* m1  = (const float*)d_in[12];
  const float* v1  = (const float*)d_in[13];
  float* out = (float*)d_out;

  int* fidx = (int*)d_ws;                 // [NB*NPOINT]            32 KB
  int* gidx = fidx + NB * NPOINT;         // [NB*NPOINT*NSAMP]       2 MB

  fps_kernel<<<NB, 1024, 0, stream>>>(xyz, fidx);
  ballq_kernel<<<NB * NPOINT, 256, 0, stream>>>(xyz, fidx, gidx);
  samlp_kernel<<<NB * NPOINT, 256, 0, stream>>>(points, gidx,
      W0, b0, g0, be0, m0, v0, W1, b1, g1, be1, m1, v1, out);
}